// EpipolarGeometryFeatureExtraction_26611617366463
// MI455X (gfx1250) — compile-verified
//
#include <hip/hip_runtime.h>

typedef float v2f __attribute__((ext_vector_type(2)));
typedef float v8f __attribute__((ext_vector_type(8)));

#define C 64  // C_IN == C_OUT == 64

__device__ __forceinline__ void atomic_add_f32(float* p, float v) {
    // Relaxed agent-scope fp32 atomic -> hardware global_atomic_add_f32 (no CAS loop)
    __hip_atomic_fetch_add(p, v, __ATOMIC_RELAXED, __HIP_MEMORY_SCOPE_AGENT);
}

__global__ void zero_f32(float* __restrict__ p, int n) {
    int t = blockIdx.x * blockDim.x + threadIdx.x;
    if (t < n) p[t] = 0.f;
}

// deg[row[e]] += ew[e]
__global__ void deg_kernel(const float* __restrict__ ew, const int* __restrict__ row,
                           float* __restrict__ deg, int E) {
    int t = blockIdx.x * blockDim.x + threadIdx.x;
    if (t < E) atomic_add_f32(&deg[row[t]], ew[t]);
}

// in-place: deg -> deg>0 ? rsqrt(deg) : 0
__global__ void dis_kernel(float* __restrict__ deg, int N) {
    int t = blockIdx.x * blockDim.x + threadIdx.x;
    if (t < N) {
        float d = deg[t];
        deg[t] = (d > 0.f) ? rsqrtf(d) : 0.f;
    }
}

// w[e] = -dis[row]*ew*dis[col]
__global__ void edgew_kernel(const float* __restrict__ dis, const float* __restrict__ ew,
                             const int* __restrict__ row, const int* __restrict__ col,
                             float* __restrict__ w, int E) {
    int t = blockIdx.x * blockDim.x + threadIdx.x;
    if (t < E) w[t] = -dis[row[t]] * ew[t] * dis[col[t]];
}

// One wave32 per edge: hout[row[e], :] += w[e] * hin[col[e], :]  (64 channels)
// Lanes cover channels [lane] and [lane+32]: 128B coalesced gather + HW fp32 atomics.
__global__ void spmm_kernel(const float* __restrict__ w, const int* __restrict__ row,
                            const int* __restrict__ col, const float* __restrict__ hin,
                            float* __restrict__ hout, int E) {
    int wv   = (blockIdx.x * blockDim.x + threadIdx.x) >> 5;
    int lane = threadIdx.x & 31;
    if (wv >= E) return;
    int r = row[wv], c = col[wv];
    float wt = w[wv];
    const float* src = hin + (long)c * C;
    float*       dst = hout + (long)r * C;
    atomic_add_f32(dst + lane,      wt * src[lane]);
    atomic_add_f32(dst + lane + 32, wt * src[lane + 32]);
}

// in-place: t2 = 2*t2 - x
__global__ void t2_kernel(float* __restrict__ t2, const float* __restrict__ x, int n) {
    int t = blockIdx.x * blockDim.x + threadIdx.x;
    if (t < n) t2[t] = 2.f * t2[t] - x[t];
}

// out = relu(T0@W0 + T1@W1 + T2@W2 + bias)
// Grid: N/16 blocks x 128 threads (4 waves). Block b handles M-tile b; wave w handles N-tile w.
// Exact tiling (N % 16 == 0) -> EXEC all-ones for every WMMA.
__global__ void cheb_gemm_kernel(const float* __restrict__ T0, const float* __restrict__ T1,
                                 const float* __restrict__ T2, const float* __restrict__ W,
                                 const float* __restrict__ bias, float* __restrict__ out) {
    const int ntile = threadIdx.x >> 5;       // 0..3
    const int mtile = blockIdx.x;             // 0..N/16-1
    const int lane  = threadIdx.x & 31;
    const int m     = lane & 15;              // row within A tile / col within B,C,D tile
    const int half  = lane >> 4;              // K-pair select (A/B), M-half select (C/D)
    const int m0    = mtile * 16;
    const int n0    = ntile * 16;

    v8f acc = {};
    const float* As[3] = { T0, T1, T2 };
#pragma unroll
    for (int k = 0; k < 3; ++k) {
        const float* A  = As[k];
        const float* Wk = W + k * C * C;
#pragma unroll
        for (int j = 0; j < 16; ++j) {        // K = 64 in steps of 4
            int kk = j * 4 + 2 * half;
            // A 16x4 tile: lane holds A[m][kk], A[m][kk+1]
            v2f a = *(const v2f*)(A + (long)(m0 + m) * C + kk);
            // B 4x16 tile (mirrored layout): lane holds W[kk][n0+m], W[kk+1][n0+m]
            v2f b;
            b.x = Wk[(kk    ) * C + n0 + m];
            b.y = Wk[(kk + 1) * C + n0 + m];
            acc = __builtin_amdgcn_wmma_f32_16x16x4_f32(
                /*neg_a=*/false, a, /*neg_b=*/false, b,
                /*c_mod=*/(short)0, acc, /*reuse_a=*/false, /*reuse_b=*/false);
        }
    }

    float bv = bias[n0 + m];
#pragma unroll
    for (int g = 0; g < 8; ++g) {             // C/D: VGPR g -> row m0 + 8*half + g
        float v = acc[g] + bv;
        v = v > 0.f ? v : 0.f;
        out[(long)(m0 + 8 * half + g) * C + n0 + m] = v;
    }
}

extern "C" void kernel_launch(void* const* d_in, const int* in_sizes, int n_in,
                              void* d_out, int out_size, void* d_ws, size_t ws_size,
                              hipStream_t stream) {
    const float* x    = (const float*)d_in[0];   // [N, 64]
    const int*   ei   = (const int*)d_in[1];     // [2, E]
    const float* ew   = (const float*)d_in[2];   // [E]
    const float* W    = (const float*)d_in[3];   // [3, 64, 64]
    const float* bias = (const float*)d_in[4];   // [64]
    float*       out  = (float*)d_out;           // [N, 64]

    const int N = in_sizes[0] / C;
    const int E = in_sizes[2];
    const int* row = ei;
    const int* col = ei + E;

    float* ws   = (float*)d_ws;
    float* deg  = ws;                          // N   (becomes dis in-place)
    float* wn   = deg + N;                     // E
    float* T1   = wn + E;                      // N*64
    float* T2   = T1 + (size_t)N * C;          // N*64

    const int B  = 256;
    const int nf = N * C;

    zero_f32<<<(N + B - 1) / B, B, 0, stream>>>(deg, N);
    deg_kernel<<<(E + B - 1) / B, B, 0, stream>>>(ew, row, deg, E);
    dis_kernel<<<(N + B - 1) / B, B, 0, stream>>>(deg, N);
    edgew_kernel<<<(E + B - 1) / B, B, 0, stream>>>(deg, ew, row, col, wn, E);

    // T1 = L_hat @ x
    zero_f32<<<(nf + B - 1) / B, B, 0, stream>>>(T1, nf);
    spmm_kernel<<<((E * 32) + B - 1) / B, B, 0, stream>>>(wn, row, col, x, T1, E);

    // T2 = 2 * (L_hat @ T1) - x
    zero_f32<<<(nf + B - 1) / B, B, 0, stream>>>(T2, nf);
    spmm_kernel<<<((E * 32) + B - 1) / B, B, 0, stream>>>(wn, row, col, T1, T2, E);
    t2_kernel<<<(nf + B - 1) / B, B, 0, stream>>>(T2, x, nf);

    // out = relu(x@W0 + T1@W1 + T2@W2 + b) via fp32 WMMA
    cheb_gemm_kernel<<<N / 16, 128, 0, stream>>>(x, T1, T2, W, bias, out);
}